// SECTRGC_1365799600308
// MI455X (gfx1250) — compile-verified
//
#include <hip/hip_runtime.h>
#include <hip/hip_bf16.h>

// Problem constants (match the reference)
#define NB 128   // batch
#define CC 64    // channels
#define TT 128   // frames
#define VV 25    // joints
#define RR 8     // rel channels
#define HH 12    // SE hidden

#define VP 28    // K padded to multiple of 4 (WMMA f32 16x16x4) -- B side only
#define UP 32    // u padded to 2x16 tiles

typedef __attribute__((ext_vector_type(2))) float v2f;
typedef __attribute__((ext_vector_type(8))) float v8f;

// ---------------------------------------------------------------------------
// Kernel 1: xm[n,c,v] = mean_t x[n,c,t,v].  One wave per (n,c); lane = v.
// ---------------------------------------------------------------------------
__global__ void reduce_xm_kernel(const float* __restrict__ x,
                                 float* __restrict__ xm) {
    int nc = blockIdx.x;            // n*CC + c
    int v  = threadIdx.x;           // 32 lanes, 25 active
    if (v >= VV) return;
    const float* p = x + (size_t)nc * (TT * VV);
    float s = 0.f;
    #pragma unroll 4
    for (int t = 0; t < TT; ++t) s += p[t * VV + v];
    xm[nc * VV + v] = s * (1.f / (float)TT);
}

// ---------------------------------------------------------------------------
// Kernel 2: per-n block. Computes x1, x2, s, SE attention a[n,v],
// rel = tanh(x1[u]-x2[v]) in LDS, then the PRE-SCALED dynamic adjacency
//   Adyn'[n,c,u,v] = (sum_r w4[c,r]*rel[r,u,v] + b4[c] + A[u,v]) * a[n,v]
// (a is folded into the small V x V side so kernel 3's x staging is a pure copy)
// ---------------------------------------------------------------------------
__global__ void build_adyn_kernel(const float* __restrict__ xm,
                                  const float* __restrict__ A,
                                  const float* __restrict__ w1, const float* __restrict__ b1,
                                  const float* __restrict__ w2, const float* __restrict__ b2,
                                  const float* __restrict__ w4, const float* __restrict__ b4,
                                  const float* __restrict__ sw1, const float* __restrict__ sb1,
                                  const float* __restrict__ sw2, const float* __restrict__ sb2,
                                  float* __restrict__ adyn) {
    __shared__ float xm_s[CC * VV];           // 1600
    __shared__ float x1_s[RR * VV];           // 200
    __shared__ float x2_s[RR * VV];           // 200
    __shared__ float s_s[VV];
    __shared__ float h_s[HH];
    __shared__ float a_s[VV];
    __shared__ float rel_s[RR * VV * VV];     // 5000 (20 KB)

    const int n = blockIdx.x;
    const int tid = threadIdx.x;              // 256 threads

    for (int i = tid; i < CC * VV; i += 256) xm_s[i] = xm[n * (CC * VV) + i];
    __syncthreads();

    // x1[r,v], x2[r,v] : projections of pooled features (R*V = 200 outputs)
    if (tid < RR * VV) {
        int r = tid / VV, v = tid - r * VV;
        float s1 = b1[r], s2 = b2[r];
        for (int c = 0; c < CC; ++c) {
            float xv = xm_s[c * VV + v];
            s1 = fmaf(w1[r * CC + c], xv, s1);
            s2 = fmaf(w2[r * CC + c], xv, s2);
        }
        x1_s[tid] = s1;
        x2_s[tid] = s2;
    }
    // s[v] = mean over c of xm  (threads 224.. overlap with x1/x2 work)
    if (tid >= 224 && tid < 224 + VV) {
        int v = tid - 224;
        float acc = 0.f;
        for (int c = 0; c < CC; ++c) acc += xm_s[c * VV + v];
        s_s[v] = acc * (1.f / (float)CC);
    }
    __syncthreads();

    // SE hidden: h = relu(sw1 @ s + sb1)
    if (tid < HH) {
        float acc = sb1[tid];
        for (int v = 0; v < VV; ++v) acc = fmaf(sw1[tid * VV + v], s_s[v], acc);
        h_s[tid] = fmaxf(acc, 0.f);
    }
    __syncthreads();

    // a = sigmoid(sw2 @ h + sb2)
    if (tid < VV) {
        float acc = sb2[tid];
        for (int j = 0; j < HH; ++j) acc = fmaf(sw2[tid * HH + j], h_s[j], acc);
        a_s[tid] = 1.f / (1.f + __expf(-acc));
    }

    // rel[r,u,v] = tanh(x1[r,u] - x2[r,v])   (x1/x2 visible since last barrier)
    for (int i = tid; i < RR * VV * VV; i += 256) {
        int r = i / (VV * VV);
        int uv = i - r * (VV * VV);
        int u = uv / VV, v = uv - u * VV;
        rel_s[i] = tanhf(x1_s[r * VV + u] - x2_s[r * VV + v]);
    }
    __syncthreads();   // rel_s and a_s now visible

    // Adyn'[n,c,u,v] = (w4.rel + b4 + A) * a[v]     (alpha == 1)
    float* dst = adyn + (size_t)n * (CC * VV * VV);
    for (int i = tid; i < CC * VV * VV; i += 256) {
        int c = i / (VV * VV);
        int uv = i - c * (VV * VV);
        int v = uv % VV;
        float acc = b4[c];
        #pragma unroll
        for (int r = 0; r < RR; ++r)
            acc = fmaf(w4[c * RR + r], rel_s[r * (VV * VV) + uv], acc);
        dst[i] = (acc + A[uv]) * a_s[v];
    }
}

// ---------------------------------------------------------------------------
// Kernel 3: out[n,c,t,u] = sum_v Adyn'[n,c,u,v] * x[n,c,t,v]
// One block per (n,c): batched (128x25)@(25x25)^T GEMM via
// V_WMMA_F32_16X16X4_F32 (fp32 -- HBM-bound workload, keep full precision).
//   - x slice (12800 contiguous bytes) staged into LDS with
//     GLOBAL_LOAD_ASYNC_TO_LDS_B128 (ASYNCcnt path), layout-preserving.
//   - Adyn' staged zero-padded on the VALU concurrently; only the B side
//     needs K-padding (A garbage at k>=25 is multiplied by B zeros).
//   - 8 waves; wave w owns t-tile 16w; 2 u-tiles x 7 K-steps.
// ---------------------------------------------------------------------------
__global__ void __launch_bounds__(256)
graph_agg_wmma_kernel(const float* __restrict__ x,
                      const float* __restrict__ adyn,
                      float* __restrict__ out) {
    __shared__ float lds_x[TT * VV + 4];   // unpadded stride-25 copy of x slice
    __shared__ float lds_ad[UP * VP];      // Adyn' zero-padded (32x28)

    const int nc  = blockIdx.x;                 // n*CC + c
    const int tid = threadIdx.x;                // 256 threads = 8 waves

    const float*   xptr  = x    + (size_t)nc * (TT * VV);
    const float*   adptr = adyn + (size_t)nc * (VV * VV);
    const uint64_t xbase = (uint64_t)(uintptr_t)xptr;
    const uint32_t ldsx  = (uint32_t)(uintptr_t)(void*)lds_x;

    // --- async copy: 800 x 16B chunks, global -> LDS, tracked by ASYNCcnt ---
    for (int i = tid; i < (TT * VV * 4) / 16; i += 256) {
        uint32_t goff = (uint32_t)i * 16u;
        uint32_t la   = ldsx + goff;
        asm volatile("global_load_async_to_lds_b128 %0, %1, %2"
                     :: "v"(la), "v"(goff), "s"(xbase)
                     : "memory");
    }

    // --- stage Adyn' (zero-padded) on the VALU while the async copy runs ---
    for (int i = tid; i < UP * VP; i += 256) {
        int u = i / VP, v = i - u * VP;
        lds_ad[i] = (u < VV && v < VV) ? adptr[u * VV + v] : 0.f;
    }

    asm volatile("s_wait_asynccnt 0x0" ::: "memory");
    __syncthreads();

    // WMMA fragment addressing (ISA 16x4 f32 A-layout, mirrored for B):
    //   lane L<16 : rows M=L,    holds K = k0+0 (vgpr0), k0+1 (vgpr1)
    //   lane L>=16: rows M=L-16, holds K = k0+2 (vgpr0), k0+3 (vgpr1)
    const int w     = tid >> 5;           // wave id 0..7 -> t-tile
    const int L     = tid & 31;
    const int row   = L & 15;             // M row (A) / N col (B, D)
    const int khalf = (L >> 4) << 1;      // 0 or 2
    const int t0    = w * 16;

    float* optr = out + (size_t)nc * (TT * VV);

    #pragma unroll
    for (int ut = 0; ut < 2; ++ut) {
        const int u0 = ut * 16;
        v8f acc = {0.f, 0.f, 0.f, 0.f, 0.f, 0.f, 0.f, 0.f};
        #pragma unroll
        for (int k0 = 0; k0 < VP; k0 += 4) {
            v2f av, bv;
            av.x = lds_x[(t0 + row) * VV + k0 + khalf];       // garbage at k>=25 OK
            av.y = lds_x[(t0 + row) * VV + k0 + khalf + 1];   // (B is zero there)
            bv.x = lds_ad[(u0 + row) * VP + k0 + khalf];
            bv.y = lds_ad[(u0 + row) * VP + k0 + khalf + 1];
            acc = __builtin_amdgcn_wmma_f32_16x16x4_f32(
                /*neg_a=*/false, av, /*neg_b=*/false, bv,
                /*c_mod=*/(short)0, acc, /*reuse_a=*/false, /*reuse_b=*/false);
        }
        // D layout: VGPR r -> row M = r + (L<16 ? 0 : 8), col N = L%16
        const int u = u0 + row;
        if (u < VV) {
            const int mbase = t0 + ((L < 16) ? 0 : 8);
            #pragma unroll
            for (int r = 0; r < 8; ++r)
                optr[(mbase + r) * VV + u] = acc[r];
        }
    }
}

// ---------------------------------------------------------------------------
extern "C" void kernel_launch(void* const* d_in, const int* in_sizes, int n_in,
                              void* d_out, int out_size, void* d_ws, size_t ws_size,
                              hipStream_t stream) {
    const float* x   = (const float*)d_in[0];
    const float* A   = (const float*)d_in[1];
    const float* w1  = (const float*)d_in[2];
    const float* b1  = (const float*)d_in[3];
    const float* w2  = (const float*)d_in[4];
    const float* b2  = (const float*)d_in[5];
    const float* w4  = (const float*)d_in[6];
    const float* b4  = (const float*)d_in[7];
    const float* sw1 = (const float*)d_in[8];
    const float* sb1 = (const float*)d_in[9];
    const float* sw2 = (const float*)d_in[10];
    const float* sb2 = (const float*)d_in[11];
    float* out = (float*)d_out;

    // Workspace partition (floats): xm (N*C*V) | Adyn' (N*C*V*V)
    float* ws_xm   = (float*)d_ws;
    float* ws_adyn = ws_xm + (size_t)NB * CC * VV;

    reduce_xm_kernel<<<NB * CC, 32, 0, stream>>>(x, ws_xm);

    build_adyn_kernel<<<NB, 256, 0, stream>>>(ws_xm, A, w1, b1, w2, b2, w4, b4,
                                              sw1, sb1, sw2, sb2, ws_adyn);

    graph_agg_wmma_kernel<<<NB * CC, 256, 0, stream>>>(x, ws_adyn, out);
}